// WeightedMSELoss_8753143349669
// MI455X (gfx1250) — compile-verified
//
#include <hip/hip_runtime.h>
#include <hip/hip_bf16.h>
#include <stdint.h>

// ---------------------------------------------------------------------------
// WeightedMSELoss (DeepVO-style): pred/target (16384,128,6) f32.
//   trans_loss = mean((p[:,:, :3]-t[:,:, :3])^2) * 1.0
//   rot_loss   = mean((wrap(p[:,:,3:])-wrap(t[:,:,3:]))^2) * 100.0
//   out = (total, trans, rot)
// Bandwidth-bound: ~100.7 MB @ 23.3 TB/s => ~4.3us floor.
// Kernel 1: async global->LDS double-buffered streaming reduction (ASYNCcnt).
// Kernel 2: V_WMMA_F32_16X16X4_F32 with all-ones B folds partials into C.
// ---------------------------------------------------------------------------

typedef __attribute__((ext_vector_type(2))) float v2f;
typedef __attribute__((ext_vector_type(8))) float v8f;
typedef __attribute__((ext_vector_type(4))) int   v4i;

#define AS1 __attribute__((address_space(1)))
#define AS3 __attribute__((address_space(3)))

#ifndef __has_builtin
#define __has_builtin(x) 0
#endif

__device__ __forceinline__ void async_copy_b128(const void* gsrc, void* lds_dst) {
#if __has_builtin(__builtin_amdgcn_global_load_async_to_lds_b128)
  __builtin_amdgcn_global_load_async_to_lds_b128(
      (AS1 v4i*)gsrc, (AS3 v4i*)lds_dst, /*offset=*/0, /*cpol=*/0);
#else
  unsigned loff = (unsigned)(size_t)(AS3 char*)lds_dst;        // LDS byte offset
  unsigned long long ga = (unsigned long long)(size_t)gsrc;    // 64-bit global addr
  asm volatile("global_load_async_to_lds_b128 %0, %1, off"
               :: "v"(loff), "v"(ga) : "memory");
#endif
}

__device__ __forceinline__ void wait_async_le2() {
#if __has_builtin(__builtin_amdgcn_s_wait_asynccnt)
  __builtin_amdgcn_s_wait_asynccnt(2);
  asm volatile("" ::: "memory");
#else
  asm volatile("s_wait_asynccnt 0x2" ::: "memory");
#endif
}

__device__ __forceinline__ void wait_async_le0() {
#if __has_builtin(__builtin_amdgcn_s_wait_asynccnt)
  __builtin_amdgcn_s_wait_asynccnt(0);
  asm volatile("" ::: "memory");
#else
  asm volatile("s_wait_asynccnt 0x0" ::: "memory");
#endif
}

__device__ __forceinline__ float wrap_pi(float a) {
  const float PI     = 3.14159274101257324f;   // float32(np.pi)
  const float TWO_PI = 6.28318548202514648f;   // float32(2*np.pi)
  float r = (a > PI) ? a - TWO_PI : a;
  r = (r < -PI) ? r + TWO_PI : r;
  return r;
}

// ---------------------------------------------------------------------------
// Kernel 1: stream both tensors through LDS via async b128 copies, accumulate
// squared diffs into (trans, rot) per-thread, reduce to per-block partials.
// Each lane reads back exactly the LDS bytes its own async op wrote, so the
// only sync needed is s_wait_asynccnt (no barriers in the hot loop).
// ---------------------------------------------------------------------------
__global__ __launch_bounds__(256)
void dvo_stream_reduce(const float4* __restrict__ pred4,
                       const float4* __restrict__ targ4,
                       float* __restrict__ partial,   // [0..grid) trans, [grid..2*grid) rot
                       long long n4) {
  __shared__ float4 buf[2][2][256];   // [stage][tensor][tid] = 16 KB
  const int tid = threadIdx.x;
  const long long stride = (long long)gridDim.x * 256;
  const long long q0 = (long long)blockIdx.x * 256 + tid;

  // Prologue: prime two pipeline stages.
  if (q0 < n4) {
    async_copy_b128(&pred4[q0], &buf[0][0][tid]);
    async_copy_b128(&targ4[q0], &buf[0][1][tid]);
  }
  if (q0 + stride < n4) {
    async_copy_b128(&pred4[q0 + stride], &buf[1][0][tid]);
    async_copy_b128(&targ4[q0 + stride], &buf[1][1][tid]);
  }

  float tacc = 0.0f, racc = 0.0f;
  int s = 0;
  for (long long q = q0; q < n4; q += stride) {
    // Outstanding tiles in order: {k, k+1?}. <=2 retires tile k's two copies.
    if (q + stride < n4) wait_async_le2(); else wait_async_le0();

    float4 p4 = buf[s][0][tid];
    float4 t4 = buf[s][1][tid];
    // Drain our ds_loads before the async engine may overwrite this stage.
    asm volatile("s_wait_dscnt 0x0" ::: "memory");

    long long q2 = q + 2 * stride;
    if (q2 < n4) {
      async_copy_b128(&pred4[q2], &buf[s][0][tid]);
      async_copy_b128(&targ4[q2], &buf[s][1][tid]);
    }
    s ^= 1;

    // Channel of flat element e is e % 6; 0..2 -> translation, 3..5 -> rotation.
    unsigned r6 = (unsigned)((4ull * (unsigned long long)q) % 6ull);
    float pv[4] = {p4.x, p4.y, p4.z, p4.w};
    float tv[4] = {t4.x, t4.y, t4.z, t4.w};
#pragma unroll
    for (int j = 0; j < 4; ++j) {
      unsigned c = r6 + (unsigned)j;
      c = (c >= 6u) ? c - 6u : c;
      bool isT = (c < 3u);
      float dtr = pv[j] - tv[j];
      float drt = wrap_pi(pv[j]) - wrap_pi(tv[j]);
      tacc = fmaf(isT ? dtr : 0.0f, dtr, tacc);
      racc = fmaf(isT ? 0.0f : drt, drt, racc);
    }
  }

  // wave32 tree reduction
#pragma unroll
  for (int off = 16; off > 0; off >>= 1) {
    tacc += __shfl_xor(tacc, off, 32);
    racc += __shfl_xor(racc, off, 32);
  }

  __shared__ float red[2][8];
  const int wave = tid >> 5, lane = tid & 31;
  if (lane == 0) { red[0][wave] = tacc; red[1][wave] = racc; }
  __syncthreads();
  if (tid == 0) {
    float ts = 0.0f, rs = 0.0f;
#pragma unroll
    for (int w = 0; w < 8; ++w) { ts += red[0][w]; rs += red[1][w]; }
    partial[blockIdx.x] = ts;
    partial[gridDim.x + blockIdx.x] = rs;
  }
}

// ---------------------------------------------------------------------------
// Kernel 2: WMMA reduction of block partials.
// D = A(16x4) x OnesB(4x16) + C  =>  D[m][n] = sum_k A[m][k] + C[m][n],
// i.e. each WMMA folds 64 fp32 partials into the accumulator (any lane->A
// bijection is valid since addition commutes). Wave0 = trans, wave1 = rot.
// Grand total per wave = sum(c[0..7]) + shfl_xor(.,16).
// ---------------------------------------------------------------------------
__global__ __launch_bounds__(64)
void dvo_finalize(const float* __restrict__ partial, int nblocks,
                  float* __restrict__ out, float inv_count) {
  const int tid = threadIdx.x;
  const int wave = tid >> 5, lane = tid & 31;
  const float* base = partial + (long long)wave * nblocks;

  v8f c = {0.f, 0.f, 0.f, 0.f, 0.f, 0.f, 0.f, 0.f};
  v2f b; b.x = 1.0f; b.y = 1.0f;              // B = ones(4x16)

  for (int i = 0; i < nblocks; i += 64) {     // nblocks multiple of 64
    v2f a;
    a.x = base[i + lane];
    a.y = base[i + 32 + lane];
    c = __builtin_amdgcn_wmma_f32_16x16x4_f32(
        /*neg_a=*/false, a, /*neg_b=*/false, b,
        /*c_mod=*/(short)0, c, /*reuse_a=*/false, /*reuse_b=*/false);
  }

  float ssum = c[0] + c[1] + c[2] + c[3] + c[4] + c[5] + c[6] + c[7];
  ssum += __shfl_xor(ssum, 16, 32);           // rows 0-7 (lanes<16) + rows 8-15

  __shared__ float sums[2];
  if (lane == 0) sums[wave] = ssum;
  __syncthreads();
  if (tid == 0) {
    float trans = sums[0] * inv_count;            // TRANS_WEIGHT = 1.0
    float rot   = sums[1] * 100.0f * inv_count;   // ROT_WEIGHT   = 100.0
    out[0] = trans + rot;
    out[1] = trans;
    out[2] = rot;
  }
}

extern "C" void kernel_launch(void* const* d_in, const int* in_sizes, int n_in,
                              void* d_out, int out_size, void* d_ws, size_t ws_size,
                              hipStream_t stream) {
  (void)n_in; (void)out_size; (void)ws_size;
  const float4* pred4 = (const float4*)d_in[0];
  const float4* targ4 = (const float4*)d_in[1];
  float* out = (float*)d_out;
  float* ws  = (float*)d_ws;

  const long long n  = (long long)in_sizes[0];  // 16384*128*6 = 12,582,912
  const long long n4 = n / 4;                   // 3,145,728 float4 tiles
  const int blocks = 2048;                      // multiple of 64 for kernel 2
  const int threads = 256;

  dvo_stream_reduce<<<blocks, threads, 0, stream>>>(pred4, targ4, ws, n4);

  const float inv_count = 1.0f / (float)(n / 2); // per-channel-group count B*T*3
  dvo_finalize<<<1, 64, 0, stream>>>(ws, blocks, out, inv_count);
}